// MotionShapeCompleteModel_62792421867849
// MI455X (gfx1250) — compile-verified
//
#include <hip/hip_runtime.h>
#include <hip/hip_bf16.h>

typedef __attribute__((ext_vector_type(2))) float v2f;
typedef __attribute__((ext_vector_type(8))) float v8f;

#define PD 64
#define PMAP_SIZE (PD * PD * PD)   // 262144 parent cells -> 1 MB int32 in d_ws
#define ROWS_PER_BLOCK 128
#define NCOLS 37                   // occ(1) + sdf(1) + mot(3) + feats(32)
#define TILE_STRIDE 41             // 37 data cols + pad -> odd stride, conflict-free A reads

// ---------------------------------------------------------------------------
// Kernel 1: clear the parent map (d_ws is poisoned; rebuild every call)
// ---------------------------------------------------------------------------
__global__ void init_pmap_kernel(int* __restrict__ pmap) {
    int i = blockIdx.x * blockDim.x + threadIdx.x;
    if (i < PMAP_SIZE) pmap[i] = -1;
}

// ---------------------------------------------------------------------------
// Kernel 2: scatter parent index at its (x,y,z) on the 64^3 parent grid.
// A query child (nx,ny,nz) on the 128^3 grid hits a scattered feature row
// iff pmap[(nx>>1, ny>>1, nz>>1)] >= 0, and that row == feats[parent].
// ---------------------------------------------------------------------------
__global__ void scatter_kernel(const int* __restrict__ locs, int N,
                               int* __restrict__ pmap) {
    int i = blockIdx.x * blockDim.x + threadIdx.x;
    if (i < N) {
        int x = locs[i * 4 + 0];
        int y = locs[i * 4 + 1];
        int z = locs[i * 4 + 2];
        pmap[(x * PD + y) * PD + z] = i;
    }
}

// ---------------------------------------------------------------------------
// Kernel 3: gather feature rows, run the 3 linear heads on the matrix pipe
// (V_WMMA_F32_16X16X4_F32 chained over K=32), stream [128,37] tiles out
// fully coalesced.
// ---------------------------------------------------------------------------
__global__ __launch_bounds__(256) void gather_heads_kernel(
    const int* __restrict__ nlocs, const float* __restrict__ feats,
    const int* __restrict__ pmap,
    const float* __restrict__ Wocc, const float* __restrict__ bocc,
    const float* __restrict__ Wsdf, const float* __restrict__ bsdf,
    const float* __restrict__ Wmot, const float* __restrict__ bmot,
    float* __restrict__ out, int M) {
    __shared__ float tile[ROWS_PER_BLOCK][TILE_STRIDE];  // cols 0-4 heads, 5-36 feats
    __shared__ float Bm[32][16];                         // packed W, cols 5-15 zero

    const int t = threadIdx.x;

    // ---- stage packed B matrix [K=32][N=16] ----
    for (int idx = t; idx < 32 * 16; idx += 256) {
        int k = idx >> 4;
        int n = idx & 15;
        float v = 0.0f;
        if (n == 0)      v = Wocc[k];
        else if (n == 1) v = Wsdf[k];
        else if (n < 5)  v = Wmot[k * 3 + (n - 2)];
        Bm[k][n] = v;
    }

    // ---- gather one feature row per thread-pair into LDS (cols 5..36) ----
    const int rowBase = blockIdx.x * ROWS_PER_BLOCK;
    const int r = t >> 1;   // local row 0..127
    const int h = t & 1;    // which 16-float half
    const int row = rowBase + r;

    float4 v0 = {0.f, 0.f, 0.f, 0.f}, v1 = v0, v2 = v0, v3 = v0;
    if (row < M) {
        const int4 L = ((const int4*)nlocs)[row];
        const int pf = (((L.x >> 1) * PD) + (L.y >> 1)) * PD + (L.z >> 1);
        const int p = pmap[pf];
        if (p >= 0) {
            const float4* src = (const float4*)(feats + (size_t)p * 32 + h * 16);
            v0 = src[0]; v1 = src[1]; v2 = src[2]; v3 = src[3];
        }
    }
    float* trow = &tile[r][5 + h * 16];
    trow[0]  = v0.x; trow[1]  = v0.y; trow[2]  = v0.z; trow[3]  = v0.w;
    trow[4]  = v1.x; trow[5]  = v1.y; trow[6]  = v1.z; trow[7]  = v1.w;
    trow[8]  = v2.x; trow[9]  = v2.y; trow[10] = v2.z; trow[11] = v2.w;
    trow[12] = v3.x; trow[13] = v3.y; trow[14] = v3.z; trow[15] = v3.w;

    __syncthreads();

    // ---- per-wave 16x16 head GEMM: D = A(16x32) * B(32x16) + bias ----
    // A frag (16x4 f32 ISA layout): lane m = lane&15 holds row M=m;
    //   VGPR0 = K=(kb), VGPR1 = K=(kb+1), kb = (lane>=16 ? 2 : 0).
    // B frag (4x16): VGPR0 lanes0-15 -> K=kb row, cols N=lane&15 (symmetric).
    // C/D (16x16): lane holds col N=lane&15; VGPR r -> M=r (+8 for lanes>=16).
    const int wave = t >> 5;
    const int lane = t & 31;
    const int mloc = lane & 15;
    const int hi   = lane >> 4;
    const int kb   = hi * 2;
    const int mBase = wave * 16;

    float bias = 0.0f;
    if (mloc == 0)      bias = bocc[0];
    else if (mloc == 1) bias = bsdf[0];
    else if (mloc < 5)  bias = bmot[mloc - 2];

    v8f c;
#pragma unroll
    for (int i = 0; i < 8; ++i) c[i] = bias;

#pragma unroll
    for (int k0 = 0; k0 < 32; k0 += 4) {
        v2f a, b;
        a.x = tile[mBase + mloc][5 + k0 + kb];
        a.y = tile[mBase + mloc][5 + k0 + kb + 1];
        b.x = Bm[k0 + kb][mloc];
        b.y = Bm[k0 + kb + 1][mloc];
        // 8 args: (neg_a, A, neg_b, B, c_mod, C, reuse_a, reuse_b)
        c = __builtin_amdgcn_wmma_f32_16x16x4_f32(
            false, a, false, b, (short)0, c, false, false);
    }

    // write head columns (N < 5) back into the tile
    if (mloc < 5) {
#pragma unroll
        for (int rr = 0; rr < 8; ++rr) {
            tile[mBase + hi * 8 + rr][mloc] = c[rr];
        }
    }

    __syncthreads();

    // ---- stream the whole [rowsHere, 37] tile out, coalesced float4 ----
    int rowsHere = M - rowBase;
    if (rowsHere > ROWS_PER_BLOCK) rowsHere = ROWS_PER_BLOCK;
    if (rowsHere <= 0) return;
    const int total = rowsHere * NCOLS;      // full block: 4736 floats (16B-mult)
    float* oblk = out + (size_t)rowBase * NCOLS;
    const int nv4 = total >> 2;
    float4* o4 = (float4*)oblk;
    for (int i = t; i < nv4; i += 256) {
        const int f0 = i * 4;
        float4 vv;
        vv.x = tile[(f0 + 0) / NCOLS][(f0 + 0) % NCOLS];
        vv.y = tile[(f0 + 1) / NCOLS][(f0 + 1) % NCOLS];
        vv.z = tile[(f0 + 2) / NCOLS][(f0 + 2) % NCOLS];
        vv.w = tile[(f0 + 3) / NCOLS][(f0 + 3) % NCOLS];
        o4[i] = vv;
    }
    for (int i = (nv4 << 2) + t; i < total; i += 256) {
        oblk[i] = tile[i / NCOLS][i % NCOLS];
    }
}

// ---------------------------------------------------------------------------
extern "C" void kernel_launch(void* const* d_in, const int* in_sizes, int n_in,
                              void* d_out, int out_size, void* d_ws, size_t ws_size,
                              hipStream_t stream) {
    const int*   locs  = (const int*)d_in[0];
    const float* feats = (const float*)d_in[1];
    const int*   nlocs = (const int*)d_in[2];
    const float* Wocc  = (const float*)d_in[3];
    const float* bocc  = (const float*)d_in[4];
    const float* Wsdf  = (const float*)d_in[5];
    const float* bsdf  = (const float*)d_in[6];
    const float* Wmot  = (const float*)d_in[7];
    const float* bmot  = (const float*)d_in[8];
    float* out = (float*)d_out;

    const int N = in_sizes[0] / 4;
    const int M = in_sizes[2] / 4;
    int* pmap = (int*)d_ws;   // 262144 ints = 1 MB

    init_pmap_kernel<<<(PMAP_SIZE + 255) / 256, 256, 0, stream>>>(pmap);
    scatter_kernel<<<(N + 255) / 256, 256, 0, stream>>>(locs, N, pmap);
    gather_heads_kernel<<<(M + ROWS_PER_BLOCK - 1) / ROWS_PER_BLOCK, 256, 0, stream>>>(
        nlocs, feats, pmap, Wocc, bocc, Wsdf, bsdf, Wmot, bmot, out, M);
}